// RIFRE_SEN_4054449127661
// MI455X (gfx1250) — compile-verified
//
#include <hip/hip_runtime.h>
#include <math.h>

// ---------------- problem constants (match reference) ----------------
#define B_  32
#define L_  512
#define D_  768
#define C_  97
#define CP_ 128   // C padded to multiple of 32 (GEMM M-tile) and of 4 (WMMA K-step)
#define NL_ 2

typedef float v2f __attribute__((ext_vector_type(2)));
typedef float v8f __attribute__((ext_vector_type(8)));

// ---------------- wave reduction helpers (wave32) ----------------
__device__ __forceinline__ float wave_sum(float s) {
#pragma unroll
  for (int o = 16; o; o >>= 1) s += __shfl_xor(s, o, 32);
  return s;
}
__device__ __forceinline__ float wave_max(float s) {
#pragma unroll
  for (int o = 16; o; o >>= 1) s = fmaxf(s, __shfl_xor(s, o, 32));
  return s;
}

// ---------------- WMMA f32 GEMM ----------------
// C[b] = A[b](MxK) @ B[b](KxN) [+ bias(b,N)] [+ add[b](MxN)]
// One wave computes a 32x64 output tile with V_WMMA_F32_16X16X4_F32:
// two A fragments (rows m0..m0+15 and m0+16..m0+31) share four B fragments,
// so each k-step issues 8 WMMAs per 8 scalar B loads + 2 A loads (b64).
// A frag (16x4 f32): lane 0-15 -> M=lane, VGPRs {K=0,K=1}; lanes 16-31 -> {K=2,K=3}.
// B frag (4x16 f32): lane 0-15 -> N=lane, VGPRs {K=0,K=1}; lanes 16-31 -> {K=2,K=3}.
// C/D (16x16 f32): VGPR j: lanes 0-15 -> M=j, lanes 16-31 -> M=j+8, N=lane&15.
// Rows rr >= Mvalid are stored as 0 (keeps padded buffers clean).
__global__ void __launch_bounds__(32)
gemm_wmma(const float* __restrict__ A, long sA,
          const float* __restrict__ Bw, long sB,
          const float* __restrict__ bias, long sBias,
          const float* __restrict__ add, long sAdd,
          float* __restrict__ Co, long sC,
          int Mvalid, int N, int K) {
  const int lane = threadIdx.x;
  const int b  = blockIdx.z;
  const int m0 = blockIdx.y << 5;
  const int n0 = blockIdx.x << 6;
  const int row = lane & 15;
  const int kk  = (lane >> 4) << 1;   // 0 or 2

  const float* Ab0 = A  + (size_t)b * sA + (size_t)(m0 + row) * K + kk;
  const float* Ab1 = Ab0 + (size_t)16 * K;
  const float* Bb  = Bw + (size_t)b * sB + n0 + row;

  v8f acc0[4] = {};
  v8f acc1[4] = {};
  for (int k0 = 0; k0 < K; k0 += 4) {
    v2f a0 = *(const v2f*)(Ab0 + k0);           // offsets even -> 8B aligned
    v2f a1 = *(const v2f*)(Ab1 + k0);
    const float* bp = Bb + (size_t)(k0 + kk) * N;
    v2f b0, b1, b2, b3;
    b0.x = bp[0];   b0.y = bp[0 + N];
    b1.x = bp[16];  b1.y = bp[16 + N];
    b2.x = bp[32];  b2.y = bp[32 + N];
    b3.x = bp[48];  b3.y = bp[48 + N];
    acc0[0] = __builtin_amdgcn_wmma_f32_16x16x4_f32(false, a0, false, b0, (short)0, acc0[0], false, false);
    acc0[1] = __builtin_amdgcn_wmma_f32_16x16x4_f32(false, a0, false, b1, (short)0, acc0[1], false, false);
    acc0[2] = __builtin_amdgcn_wmma_f32_16x16x4_f32(false, a0, false, b2, (short)0, acc0[2], false, false);
    acc0[3] = __builtin_amdgcn_wmma_f32_16x16x4_f32(false, a0, false, b3, (short)0, acc0[3], false, false);
    acc1[0] = __builtin_amdgcn_wmma_f32_16x16x4_f32(false, a1, false, b0, (short)0, acc1[0], false, false);
    acc1[1] = __builtin_amdgcn_wmma_f32_16x16x4_f32(false, a1, false, b1, (short)0, acc1[1], false, false);
    acc1[2] = __builtin_amdgcn_wmma_f32_16x16x4_f32(false, a1, false, b2, (short)0, acc1[2], false, false);
    acc1[3] = __builtin_amdgcn_wmma_f32_16x16x4_f32(false, a1, false, b3, (short)0, acc1[3], false, false);
  }

  const int rofs = (lane >> 4) << 3;  // 0 or 8
#pragma unroll
  for (int h = 0; h < 2; ++h) {
#pragma unroll
    for (int t = 0; t < 4; ++t) {
#pragma unroll
      for (int j = 0; j < 8; ++j) {
        int rr = m0 + (h << 4) + rofs + j;
        int cc = n0 + (t << 4) + row;
        float v = h ? acc1[t][j] : acc0[t][j];
        if (bias) v += bias[(size_t)b * sBias + cc];
        if (add)  v += add[(size_t)b * sAdd + (size_t)rr * N + cc];
        Co[(size_t)b * sC + (size_t)rr * N + cc] = (rr < Mvalid) ? v : 0.0f;
      }
    }
  }
}

// ---------------- small elementwise / reduction kernels ----------------
__global__ void copy_f32(const float* __restrict__ src, float* __restrict__ dst, long n) {
  long i = (long)blockIdx.x * blockDim.x + threadIdx.x;
  if (i < n) dst[i] = src[i];
}

// emb (97x768) -> padded (128x768), pad rows zero (same flat layout for r<97)
__global__ void pademb_k(const float* __restrict__ emb, float* __restrict__ dst) {
  int i = blockIdx.x * blockDim.x + threadIdx.x;
  if (i < CP_ * D_) dst[i] = (i < C_ * D_) ? emb[i] : 0.0f;
}

// out[b*ldOut + m] = dot(A[b][m][:D], w)
__global__ void __launch_bounds__(32)
rowdot_k(const float* __restrict__ A, long sA,
         const float* __restrict__ w, float* __restrict__ out, int ldOut) {
  int b = blockIdx.y, m = blockIdx.x, lane = threadIdx.x;
  const float* r = A + (size_t)b * sA + (size_t)m * D_;
  float s = 0.0f;
  for (int d = lane; d < D_; d += 32) s += r[d] * w[d];
  s = wave_sum(s);
  if (lane == 0) out[(size_t)b * ldOut + m] = s;
}

// softmax over keys j for one query row m: score = qs[m] + ks[j] + sb, optional mask
__global__ void __launch_bounds__(32)
softmax_k(const float* __restrict__ qs, int ldq,
          const float* __restrict__ ks, int ldk,
          const float* __restrict__ sbp, int sbIdx,
          const int* __restrict__ mask,
          float* __restrict__ attn, long batchStride, int rowStride,
          int Kr, int Kp) {
  int b = blockIdx.y, m = blockIdx.x, lane = threadIdx.x;
  float qv = qs[(size_t)b * ldq + m] + sbp[sbIdx];
  float* arow = attn + (size_t)b * batchStride + (size_t)m * rowStride;
  const int* mrow = mask ? (mask + (size_t)b * Kr) : (const int*)nullptr;
  const float* krow = ks + (size_t)b * ldk;

  float mx = -3.4e38f;
  for (int j = lane; j < Kr; j += 32) {
    float s = qv + krow[j];
    if (mrow && mrow[j] == 0) s = -1e9f;
    mx = fmaxf(mx, s);
  }
  mx = wave_max(mx);

  float sum = 0.0f;
  for (int j = lane; j < Kr; j += 32) {
    float s = qv + krow[j];
    if (mrow && mrow[j] == 0) s = -1e9f;
    float e = expf(s - mx);
    arow[j] = e;
    sum += e;
  }
  sum = wave_sum(sum);
  float inv = 1.0f / sum;
  for (int j = lane; j < Kr; j += 32) arow[j] *= inv;
  for (int j = Kr + lane; j < Kp; j += 32) arow[j] = 0.0f;  // zero pad columns
}

// fused gate + both residuals: g = sigmoid(out.gwa + X.gwb + gb); X = g*out + (2-g)*X
__global__ void __launch_bounds__(32)
gate_k(float* __restrict__ X, long sX, const float* __restrict__ out, long sO,
       const float* __restrict__ gwa, const float* __restrict__ gwb,
       const float* __restrict__ gbp, int gbIdx) {
  int b = blockIdx.y, m = blockIdx.x, lane = threadIdx.x;
  float* xr = X + (size_t)b * sX + (size_t)m * D_;
  const float* orow = out + (size_t)b * sO + (size_t)m * D_;
  float s = 0.0f;
  for (int d = lane; d < D_; d += 32) s += orow[d] * gwa[d] + xr[d] * gwb[d];
  s = wave_sum(s);
  float g = 1.0f / (1.0f + expf(-(s + gbp[gbIdx])));
  for (int d = lane; d < D_; d += 32) xr[d] = g * orow[d] + (2.0f - g) * xr[d];
}

// e[b][d] = sum_l x[b][l][d]*pos[b][l] / sum_l pos[b][l]
__global__ void __launch_bounds__(256)
pool_k(const float* __restrict__ x, const float* __restrict__ pos, float* __restrict__ e) {
  int b = blockIdx.y;
  int d = blockIdx.x * 256 + threadIdx.x;
  const float* xb = x + (size_t)b * L_ * D_ + d;
  const float* pb = pos + (size_t)b * L_;
  float s = 0.0f, cnt = 0.0f;
  for (int l = 0; l < L_; ++l) { float w = pb[l]; s += xb[(size_t)l * D_] * w; cnt += w; }
  e[(size_t)b * D_ + d] = s / cnt;
}

// r[b][n] = fc1_b[n] + e1[b]@fc1_w[D:2D] + e2[b]@fc1_w[2D:3D]  (per-batch bias for fc1 GEMM)
__global__ void __launch_bounds__(256)
rvec_k(const float* __restrict__ e1, const float* __restrict__ e2,
       const float* __restrict__ fc1w, const float* __restrict__ fc1b,
       float* __restrict__ r) {
  int b = blockIdx.y;
  int n = blockIdx.x * 256 + threadIdx.x;
  const float* w1 = fc1w + (size_t)D_ * D_ + n;
  const float* w2 = fc1w + (size_t)2 * D_ * D_ + n;
  const float* e1b = e1 + (size_t)b * D_;
  const float* e2b = e2 + (size_t)b * D_;
  float acc = fc1b[n];
  for (int d = 0; d < D_; ++d)
    acc += e1b[d] * w1[(size_t)d * D_] + e2b[d] * w2[(size_t)d * D_];
  r[(size_t)b * D_ + n] = acc;
}

// logits[b][c] = sigmoid(tanh(h[b][c]).fc2_w + fc2_b)
__global__ void __launch_bounds__(32)
head_k(const float* __restrict__ h, const float* __restrict__ fc2w,
       const float* __restrict__ fc2b, float* __restrict__ logits) {
  int b = blockIdx.y, c = blockIdx.x, lane = threadIdx.x;
  const float* hr = h + ((size_t)b * CP_ + c) * D_;
  float s = 0.0f;
  for (int d = lane; d < D_; d += 32) s += tanhf(hr[d]) * fc2w[d];
  s = wave_sum(s);
  if (lane == 0) logits[(size_t)b * C_ + c] = 1.0f / (1.0f + expf(-(s + fc2b[0])));
}

__global__ void __launch_bounds__(32)
argmax_k(const float* __restrict__ logits, float* __restrict__ pred) {
  int b = threadIdx.x;
  if (b < B_) {
    const float* lr = logits + (size_t)b * C_;
    float best = lr[0]; int bi = 0;
    for (int c = 1; c < C_; ++c) { float v = lr[c]; if (v > best) { best = v; bi = c; } }
    pred[b] = (float)bi;
  }
}

// ---------------- host orchestration ----------------
static void launch_gemm(hipStream_t st, const float* A, long sA, const float* Bw, long sB,
                        const float* bias, long sBias, const float* add, long sAdd,
                        float* Co, long sC, int M, int Mvalid, int N, int K, int batch) {
  dim3 g(N / 64, M / 32, batch);
  hipLaunchKernelGGL(gemm_wmma, g, dim3(32), 0, st,
                     A, sA, Bw, sB, bias, sBias, add, sAdd, Co, sC, Mvalid, N, K);
}

extern "C" void kernel_launch(void* const* d_in, const int* in_sizes, int n_in,
                              void* d_out, int out_size, void* d_ws, size_t ws_size,
                              hipStream_t stream) {
  const float* x_in  = (const float*)d_in[0];
  const float* pos1  = (const float*)d_in[1];
  const float* pos2  = (const float*)d_in[2];
  const int*   mask  = (const int*)d_in[3];
  const float* emb   = (const float*)d_in[4];
  const float* rel_w = (const float*)d_in[5];
  const float* rel_b = (const float*)d_in[6];
  const float* qw = (const float*)d_in[7];
  const float* qb = (const float*)d_in[8];
  const float* kw = (const float*)d_in[9];
  const float* kb = (const float*)d_in[10];
  const float* vw = (const float*)d_in[11];
  const float* vb = (const float*)d_in[12];
  const float* sw = (const float*)d_in[13];
  const float* sb = (const float*)d_in[14];
  const float* gw = (const float*)d_in[15];
  const float* gb = (const float*)d_in[16];
  const float* fc1w = (const float*)d_in[17];
  const float* fc1b = (const float*)d_in[18];
  const float* fc2w = (const float*)d_in[19];
  const float* fc2b = (const float*)d_in[20];
  (void)in_sizes; (void)n_in; (void)out_size; (void)ws_size;

  float* ws = (float*)d_ws;
  size_t o = 0;
  float* X    = ws + o; o += (size_t)B_ * L_  * D_;   // working copy of x
  float* P    = ws + o; o += (size_t)B_ * CP_ * D_;   // padded p
  float* bigA = ws + o; o += (size_t)B_ * L_  * D_;
  float* bigB = ws + o; o += (size_t)B_ * L_  * D_;
  float* s1   = ws + o; o += (size_t)B_ * CP_ * D_;
  float* s2   = ws + o; o += (size_t)B_ * CP_ * D_;
  float* attn = ws + o; o += (size_t)B_ * L_  * CP_;  // reused for both RA shapes
  float* embp = ws + o; o += (size_t)CP_ * D_;
  float* qsB  = ws + o; o += (size_t)B_ * L_;
  float* ksB  = ws + o; o += (size_t)B_ * L_;
  float* e1   = ws + o; o += (size_t)B_ * D_;
  float* e2   = ws + o; o += (size_t)B_ * D_;
  float* rv   = ws + o; o += (size_t)B_ * D_;

  long n = (long)B_ * L_ * D_;
  hipLaunchKernelGGL(copy_f32, dim3((unsigned)((n + 255) / 256)), dim3(256), 0, stream, x_in, X, n);
  hipLaunchKernelGGL(pademb_k, dim3((CP_ * D_ + 255) / 256), dim3(256), 0, stream, emb, embp);

  // p0 = emb @ rel_w + rel_b, broadcast over batch (strideA=0, strideC per batch)
  launch_gemm(stream, embp, 0, rel_w, 0, rel_b, 0, nullptr, 0,
              P, (long)CP_ * D_, CP_, C_, D_, D_, B_);

  const size_t DD = (size_t)D_ * D_;
  for (int li = 0; li < NL_; ++li) {
    int i0 = li * 2 + 0, i1 = li * 2 + 1;

    // ---- RA1: queries X (L rows), keys/values P (C rows), no mask ----
    launch_gemm(stream, X, (long)L_ * D_, qw + i0 * DD, 0, qb + (size_t)i0 * D_, 0, nullptr, 0,
                bigA, (long)L_ * D_, L_, L_, D_, D_, B_);
    launch_gemm(stream, P, (long)CP_ * D_, kw + i0 * DD, 0, kb + (size_t)i0 * D_, 0, nullptr, 0,
                s1, (long)CP_ * D_, CP_, C_, D_, D_, B_);
    launch_gemm(stream, P, (long)CP_ * D_, vw + i0 * DD, 0, vb + (size_t)i0 * D_, 0, nullptr, 0,
                s2, (long)CP_ * D_, CP_, C_, D_, D_, B_);
    hipLaunchKernelGGL(rowdot_k, dim3(L_, B_), dim3(32), 0, stream,
                       bigA, (long)L_ * D_, sw + (size_t)i0 * 2 * D_, qsB, L_);
    hipLaunchKernelGGL(rowdot_k, dim3(C_, B_), dim3(32), 0, stream,
                       s1, (long)CP_ * D_, sw + (size_t)i0 * 2 * D_ + D_, ksB, CP_);
    hipLaunchKernelGGL(softmax_k, dim3(L_, B_), dim3(32), 0, stream,
                       qsB, L_, ksB, CP_, sb, i0, (const int*)nullptr,
                       attn, (long)L_ * CP_, CP_, C_, CP_);
    launch_gemm(stream, attn, (long)L_ * CP_, s2, (long)CP_ * D_, nullptr, 0, X, (long)L_ * D_,
                bigB, (long)L_ * D_, L_, L_, D_, CP_, B_);
    hipLaunchKernelGGL(gate_k, dim3(L_, B_), dim3(32), 0, stream,
                       X, (long)L_ * D_, bigB, (long)L_ * D_,
                       gw + (size_t)i0 * 2 * D_, gw + (size_t)i0 * 2 * D_ + D_, gb, i0);

    // ---- RA2: queries P (C rows), keys/values X (L rows), masked ----
    launch_gemm(stream, P, (long)CP_ * D_, qw + i1 * DD, 0, qb + (size_t)i1 * D_, 0, nullptr, 0,
                s1, (long)CP_ * D_, CP_, C_, D_, D_, B_);
    launch_gemm(stream, X, (long)L_ * D_, kw + i1 * DD, 0, kb + (size_t)i1 * D_, 0, nullptr, 0,
                bigA, (long)L_ * D_, L_, L_, D_, D_, B_);
    launch_gemm(stream, X, (long)L_ * D_, vw + i1 * DD, 0, vb + (size_t)i1 * D_, 0, nullptr, 0,
                bigB, (long)L_ * D_, L_, L_, D_, D_, B_);
    hipLaunchKernelGGL(rowdot_k, dim3(C_, B_), dim3(32), 0, stream,
                       s1, (long)CP_ * D_, sw + (size_t)i1 * 2 * D_, qsB, CP_);
    hipLaunchKernelGGL(rowdot_k, dim3(L_, B_), dim3(32), 0, stream,
                       bigA, (long)L_ * D_, sw + (size_t)i1 * 2 * D_ + D_, ksB, L_);
    hipLaunchKernelGGL(softmax_k, dim3(C_, B_), dim3(32), 0, stream,
                       qsB, CP_, ksB, L_, sb, i1, mask,
                       attn, (long)CP_ * L_, L_, L_, L_);
    launch_gemm(stream, attn, (long)CP_ * L_, bigB, (long)L_ * D_, nullptr, 0, P, (long)CP_ * D_,
                s2, (long)CP_ * D_, CP_, C_, D_, L_, B_);
    hipLaunchKernelGGL(gate_k, dim3(C_, B_), dim3(32), 0, stream,
                       P, (long)CP_ * D_, s2, (long)CP_ * D_,
                       gw + (size_t)i1 * 2 * D_, gw + (size_t)i1 * 2 * D_ + D_, gb, i1);
  }

  // pooling + head
  hipLaunchKernelGGL(pool_k, dim3(D_ / 256, B_), dim3(256), 0, stream, X, pos1, e1);
  hipLaunchKernelGGL(pool_k, dim3(D_ / 256, B_), dim3(256), 0, stream, X, pos2, e2);
  hipLaunchKernelGGL(rvec_k, dim3(D_ / 256, B_), dim3(256), 0, stream, e1, e2, fc1w, fc1b, rv);
  // h_pre = p @ fc1_w[:D] + r[b]  (per-batch bias vector)
  launch_gemm(stream, P, (long)CP_ * D_, fc1w, 0, rv, (long)D_, nullptr, 0,
              s1, (long)CP_ * D_, CP_, C_, D_, D_, B_);

  float* logits = (float*)d_out;
  hipLaunchKernelGGL(head_k, dim3(C_, B_), dim3(32), 0, stream, s1, fc2w, fc2b, logits);
  hipLaunchKernelGGL(argmax_k, dim3(1), dim3(32), 0, stream, logits, logits + (size_t)B_ * C_);
}